// VSSBlock_89172111000240
// MI455X (gfx1250) — compile-verified
//
#include <hip/hip_runtime.h>
#include <hip/hip_bf16.h>

typedef __attribute__((ext_vector_type(2))) float v2f;
typedef __attribute__((ext_vector_type(8))) float v8f;

#define LDS_STRIDE 17   // 16 K-values + 1 pad (bank-conflict avoidance)

// ---------------------------------------------------------------------------
// LayerNorm: one block (256 threads) per row of length D
// ---------------------------------------------------------------------------
__global__ __launch_bounds__(256)
void layernorm_k(const float* __restrict__ x, const float* __restrict__ g,
                 const float* __restrict__ b, float* __restrict__ o, int D)
{
    const int row = blockIdx.x;
    const float* xr = x + (size_t)row * D;
    float s = 0.f, s2 = 0.f;
    for (int i = threadIdx.x; i < D; i += blockDim.x) {
        float v = xr[i];
        s += v;
        s2 = fmaf(v, v, s2);
    }
    for (int off = 16; off > 0; off >>= 1) {
        s  += __shfl_xor(s,  off, 32);
        s2 += __shfl_xor(s2, off, 32);
    }
    __shared__ float wsum[8], wsum2[8];
    const int wave = threadIdx.x >> 5, lane = threadIdx.x & 31;
    if (lane == 0) { wsum[wave] = s; wsum2[wave] = s2; }
    __syncthreads();
    __shared__ float mv[2];
    if (wave == 0) {
        float a  = (lane < 8) ? wsum[lane]  : 0.f;
        float a2 = (lane < 8) ? wsum2[lane] : 0.f;
        for (int off = 4; off > 0; off >>= 1) {
            a  += __shfl_xor(a,  off, 32);
            a2 += __shfl_xor(a2, off, 32);
        }
        if (lane == 0) {
            float mu  = a / (float)D;
            float var = a2 / (float)D - mu * mu;
            mv[0] = mu;
            mv[1] = rsqrtf(var + 1e-5f);
        }
    }
    __syncthreads();
    const float mu = mv[0], inv = mv[1];
    float* orow = o + (size_t)row * D;
    for (int i = threadIdx.x; i < D; i += blockDim.x) {
        float v = xr[i];
        orow[i] = fmaf((v - mu) * inv, g[i], b[i]);
    }
}

// ---------------------------------------------------------------------------
// NT GEMM:  C[M,N] = A[M,K] * W[N,K]^T (+ bias[n]) (+ res[m,n])
// 128x128 block tile, K-chunk 16 staged in LDS, 8 waves, wave tile 32x64.
// OOB n rows: address-clamped loads (results discarded by epilogue guard) —
// the hot loop is branch-free: ds_load + v_wmma only.
// ---------------------------------------------------------------------------
__global__ __launch_bounds__(256)
void gemm_nt_wmma(const float* __restrict__ A, int lda,
                  const float* __restrict__ W, int ldw,
                  const float* __restrict__ bias,
                  const float* __restrict__ res,
                  float* __restrict__ C, int ldc,
                  int M, int N, int K)
{
    __shared__ float sA[128 * LDS_STRIDE];
    __shared__ float sB[128 * LDS_STRIDE];

    const int tid  = threadIdx.x;
    const int lane = tid & 31;
    const int wave = tid >> 5;
    const int l16  = lane & 15;
    const int half = lane >> 4;
    const int wm   = wave & 3;
    const int wn   = wave >> 2;
    const int mblk = blockIdx.y * 128;
    const int nblk = blockIdx.x * 128;

    // cooperative fill mapping: thread -> (row 0..127, k-half 0/8)
    const int fr = tid >> 1;
    const int fk = (tid & 1) * 8;
    const float* gA = A + (size_t)(mblk + fr) * lda + fk;
    const int wrow = nblk + fr;
    const float* gW = W + (size_t)(wrow < N ? wrow : (N - 1)) * ldw + fk;

    v8f acc[2][4];
#pragma unroll
    for (int mt = 0; mt < 2; ++mt)
#pragma unroll
        for (int nt = 0; nt < 4; ++nt)
            acc[mt][nt] = (v8f)(0.f);

    int aoff[2], boff[4];
#pragma unroll
    for (int mt = 0; mt < 2; ++mt)
        aoff[mt] = (wm * 32 + mt * 16 + l16) * LDS_STRIDE + half * 2;
#pragma unroll
    for (int nt = 0; nt < 4; ++nt)
        boff[nt] = (wn * 64 + nt * 16 + l16) * LDS_STRIDE + half * 2;

    for (int kc = 0; kc < K; kc += 16) {
        __syncthreads();
        if (kc + 16 < K) {
            __builtin_prefetch(gA + kc + 16, 0, 0);
            __builtin_prefetch(gW + kc + 16, 0, 0);
        }
        {
            float4 a0 = *(const float4*)(gA + kc);
            float4 a1 = *(const float4*)(gA + kc + 4);
            float4 w0 = *(const float4*)(gW + kc);
            float4 w1 = *(const float4*)(gW + kc + 4);
            float* pa = &sA[fr * LDS_STRIDE + fk];
            float* pb = &sB[fr * LDS_STRIDE + fk];
            pa[0] = a0.x; pa[1] = a0.y; pa[2] = a0.z; pa[3] = a0.w;
            pa[4] = a1.x; pa[5] = a1.y; pa[6] = a1.z; pa[7] = a1.w;
            pb[0] = w0.x; pb[1] = w0.y; pb[2] = w0.z; pb[3] = w0.w;
            pb[4] = w1.x; pb[5] = w1.y; pb[6] = w1.z; pb[7] = w1.w;
        }
        __syncthreads();

#pragma unroll
        for (int kk = 0; kk < 16; kk += 4) {
            v2f af[2], bf[4];
#pragma unroll
            for (int mt = 0; mt < 2; ++mt) {
                af[mt].x = sA[aoff[mt] + kk];
                af[mt].y = sA[aoff[mt] + kk + 1];
            }
#pragma unroll
            for (int nt = 0; nt < 4; ++nt) {
                bf[nt].x = sB[boff[nt] + kk];
                bf[nt].y = sB[boff[nt] + kk + 1];
            }
#pragma unroll
            for (int mt = 0; mt < 2; ++mt)
#pragma unroll
                for (int nt = 0; nt < 4; ++nt)
                    acc[mt][nt] = __builtin_amdgcn_wmma_f32_16x16x4_f32(
                        false, af[mt], false, bf[nt], (short)0, acc[mt][nt],
                        false, false);
        }
    }

#pragma unroll
    for (int mt = 0; mt < 2; ++mt)
#pragma unroll
        for (int nt = 0; nt < 4; ++nt) {
            int n = nblk + wn * 64 + nt * 16 + l16;
            if (n < N) {
                float bn = bias ? bias[n] : 0.f;
#pragma unroll
                for (int i = 0; i < 8; ++i) {
                    int m = mblk + wm * 32 + mt * 16 + half * 8 + i;
                    size_t idx = (size_t)m * ldc + n;
                    float v = acc[mt][nt][i] + bn;
                    if (res) v += res[idx];
                    C[idx] = v;
                }
            }
        }
}

// ---------------------------------------------------------------------------
// NN GEMM (batched, shifted columns with zero pad, strided A-K):
//   C[z][M,N] = sum_k A[z][m*lda + k*aks] * B[z][k*ldb + (n+nshift)]
//              (+ biasM[m]) (+ res[z][m,n])
// Zero-padding is applied ONCE at LDS-fill time (branch-free hot loop).
// ---------------------------------------------------------------------------
__global__ __launch_bounds__(256)
void gemm_nn_wmma(const float* __restrict__ A, int lda, int aks, long long batA,
                  const float* __restrict__ B, int ldb, long long batB, int nshift,
                  const float* __restrict__ biasM,
                  const float* __restrict__ res, long long batR,
                  float* __restrict__ C, int ldc, long long batC,
                  int M, int N, int K)
{
    __shared__ float sA[128 * LDS_STRIDE];
    __shared__ float sB[128 * LDS_STRIDE];

    const int z = blockIdx.z;
    A += (size_t)z * batA;
    B += (size_t)z * batB;
    C += (size_t)z * batC;
    const float* resz = res ? res + (size_t)z * batR : nullptr;

    const int tid  = threadIdx.x;
    const int lane = tid & 31;
    const int wave = tid >> 5;
    const int l16  = lane & 15;
    const int half = lane >> 4;
    const int wm   = wave & 3;
    const int wn   = wave >> 2;
    const int mblk = blockIdx.y * 128;
    const int nblk = blockIdx.x * 128;

    // A fill mapping: thread -> (row, k-half)
    const int fr = tid >> 1;
    const int fk = (tid & 1) * 8;
    const float* gA = A + (size_t)(mblk + fr) * lda + (size_t)fk * aks;

    // B fill mapping: thread -> (col 0..127, k-half 0/8); coalesced over cols
    const int fc  = tid & 127;
    const int fk2 = (tid >> 7) * 8;
    const int col = nblk + fc + nshift;
    const float bmask = (col >= 0 && col < N) ? 1.f : 0.f;
    const int colc = (col < 0) ? 0 : (col >= N ? N - 1 : col);
    const float* gB = B + colc + (size_t)fk2 * ldb;

    v8f acc[2][4];
#pragma unroll
    for (int mt = 0; mt < 2; ++mt)
#pragma unroll
        for (int nt = 0; nt < 4; ++nt)
            acc[mt][nt] = (v8f)(0.f);

    int aoff[2], boff[4];
#pragma unroll
    for (int mt = 0; mt < 2; ++mt)
        aoff[mt] = (wm * 32 + mt * 16 + l16) * LDS_STRIDE + half * 2;
#pragma unroll
    for (int nt = 0; nt < 4; ++nt)
        boff[nt] = (wn * 64 + nt * 16 + l16) * LDS_STRIDE + half * 2;

    for (int kc = 0; kc < K; kc += 16) {
        __syncthreads();
        if (kc + 16 < K) {
            __builtin_prefetch(gA + (size_t)(kc + 16) * aks, 0, 0);
            __builtin_prefetch(gB + (size_t)(kc + 16) * ldb, 0, 0);
        }
        {
            float* pa = &sA[fr * LDS_STRIDE + fk];
#pragma unroll
            for (int j = 0; j < 8; ++j)
                pa[j] = gA[(size_t)(kc + j) * aks];
            float* pb = &sB[fc * LDS_STRIDE + fk2];
#pragma unroll
            for (int j = 0; j < 8; ++j)
                pb[j] = gB[(size_t)(kc + j) * ldb] * bmask;
        }
        __syncthreads();

#pragma unroll
        for (int kk = 0; kk < 16; kk += 4) {
            v2f af[2], bf[4];
#pragma unroll
            for (int mt = 0; mt < 2; ++mt) {
                af[mt].x = sA[aoff[mt] + kk];
                af[mt].y = sA[aoff[mt] + kk + 1];
            }
#pragma unroll
            for (int nt = 0; nt < 4; ++nt) {
                bf[nt].x = sB[boff[nt] + kk];
                bf[nt].y = sB[boff[nt] + kk + 1];
            }
#pragma unroll
            for (int mt = 0; mt < 2; ++mt)
#pragma unroll
                for (int nt = 0; nt < 4; ++nt)
                    acc[mt][nt] = __builtin_amdgcn_wmma_f32_16x16x4_f32(
                        false, af[mt], false, bf[nt], (short)0, acc[mt][nt],
                        false, false);
        }
    }

#pragma unroll
    for (int mt = 0; mt < 2; ++mt)
#pragma unroll
        for (int nt = 0; nt < 4; ++nt) {
            int n = nblk + wn * 64 + nt * 16 + l16;
            if (n < N) {
#pragma unroll
                for (int i = 0; i < 8; ++i) {
                    int m = mblk + wm * 32 + mt * 16 + half * 8 + i;
                    size_t idx = (size_t)m * ldc + n;
                    float v = acc[mt][nt][i];
                    if (biasM) v += biasM[m];
                    if (resz)  v += resz[idx];
                    C[idx] = v;
                }
            }
        }
}

// ---------------------------------------------------------------------------
// Selective scan: one thread per (b, e, n); 16 n-lanes per half-wave share a
// (b,e) channel; 4-step shfl_xor reduction over n gives y[b,l,e].
// ---------------------------------------------------------------------------
__global__ __launch_bounds__(256)
void ssm_scan_k(const float* __restrict__ xin,   // (B*L, 1024) = conv output
                const float* __restrict__ dbc,   // (B*L, 96): [64:80)=Bm [80:96)=Cm
                const float* __restrict__ dpre,  // (B*L, 1024) pre-softplus delta
                const float* __restrict__ A_log, // (1024, 16)
                const float* __restrict__ Dp,    // (1024)
                float* __restrict__ y)           // (B*L, 1024)
{
    const int tid  = blockIdx.x * blockDim.x + threadIdx.x;
    const int lane = tid & 31;
    const int n    = lane & 15;
    const int pair = (tid >> 5) * 2 + (lane >> 4);  // 0..4095  (b*1024 + e)
    const int b    = pair >> 10;
    const int e    = pair & 1023;

    const float An = -__expf(A_log[e * 16 + n]);
    const float De = Dp[e];
    float h = 0.f;
    const size_t rowbase = (size_t)b * 1024;
    for (int l = 0; l < 1024; ++l) {
        const size_t r = rowbase + l;
        float dp = dpre[r * 1024 + e];
        float sp = (dp > 20.f) ? dp : log1pf(__expf(dp));   // softplus
        float xv = xin[r * 1024 + e];
        float Bv = dbc[r * 96 + 64 + n];
        float Cv = dbc[r * 96 + 80 + n];
        float dA = __expf(sp * An);
        h = fmaf(dA, h, sp * Bv * xv);
        float t = h * Cv;
        t += __shfl_xor(t, 1, 32);
        t += __shfl_xor(t, 2, 32);
        t += __shfl_xor(t, 4, 32);
        t += __shfl_xor(t, 8, 32);
        if (n == 0) y[r * 1024 + e] = fmaf(De, xv, t);
    }
}

// ---------------------------------------------------------------------------
extern "C" void kernel_launch(void* const* d_in, const int* in_sizes, int n_in,
                              void* d_out, int out_size, void* d_ws, size_t ws_size,
                              hipStream_t stream)
{
    const float* x     = (const float*)d_in[0];   // (4,1024,1024)
    const float* ln_g  = (const float*)d_in[1];   // (1024)
    const float* ln_b  = (const float*)d_in[2];   // (1024)
    const float* Wp    = (const float*)d_in[3];   // (1024,1024)
    const float* bp    = (const float*)d_in[4];   // (1024)
    const float* Wc    = (const float*)d_in[5];   // (1024,1024,3)
    const float* bc    = (const float*)d_in[6];   // (1024)
    const float* A_log = (const float*)d_in[7];   // (1024,16)
    const float* Dp    = (const float*)d_in[8];   // (1024)
    const float* W_dbc = (const float*)d_in[9];   // (96,1024)
    const float* W_dt  = (const float*)d_in[10];  // (1024,64)
    const float* b_dt  = (const float*)d_in[11];  // (1024)
    float* out = (float*)d_out;

    const int Bn = 4, L = 1024, D = 1024, NST = 16, RANK = 64, DBC = 96;
    const int ML = Bn * L;              // 4096 rows
    const size_t SZ = (size_t)ML * D;   // 4M floats per activation buffer
    const long long BAT = (long long)L * D;

    float* ws     = (float*)d_ws;
    float* ln1    = ws;                 // slot 0
    float* h1     = ws + SZ;            // slot 1
    float* lskip  = ws + 2 * SZ;        // slot 2
    float* hconv  = ws + 3 * SZ;        // slot 3
    float* delta  = ws + 4 * SZ;        // slot 4
    float* dbcbuf = ws + 5 * SZ;        // 4096*96 floats
    float* yssm   = ln1;                // reuse slot 0 (ln1 dead after step 2)
    float* ln2    = h1;                 // reuse slot 1 (h1 dead after conv)
    float* hmm    = hconv;              // reuse slot 3 (hconv dead after scan)

    dim3 blk(256);
    dim3 g_nt(D / 128, ML / 128);           // 8 x 32
    dim3 g_nn(D / 128, D / 128, Bn);        // 8 x 8 x 4
    dim3 g_dbc((DBC + 127) / 128, ML / 128);

    // 1) ln1 = LayerNorm(x)
    layernorm_k<<<ML, blk, 0, stream>>>(x, ln_g, ln_b, ln1, D);
    // 2) h1 = ln1 @ Wp^T + bp
    gemm_nt_wmma<<<g_nt, blk, 0, stream>>>(ln1, D, Wp, D, bp, nullptr, h1, D, ML, D, D);
    // 3) lskip = h1 @ Wp^T + bp
    gemm_nt_wmma<<<g_nt, blk, 0, stream>>>(h1, D, Wp, D, bp, nullptr, lskip, D, ML, D, D);
    // 4) conv1d over feature axis as 3 shifted dense GEMMs (batched over b)
    gemm_nn_wmma<<<g_nn, blk, 0, stream>>>(Wc + 0, 3 * D, 3, 0, h1, D, BAT, -1,
                                           bc, nullptr, 0, hconv, D, BAT, D, D, D);
    gemm_nn_wmma<<<g_nn, blk, 0, stream>>>(Wc + 1, 3 * D, 3, 0, h1, D, BAT, 0,
                                           nullptr, hconv, BAT, hconv, D, BAT, D, D, D);
    gemm_nn_wmma<<<g_nn, blk, 0, stream>>>(Wc + 2, 3 * D, 3, 0, h1, D, BAT, 1,
                                           nullptr, hconv, BAT, hconv, D, BAT, D, D, D);
    // 5) dbc = hconv @ W_dbc^T   (N=96: delta_in | Bm | Cm)
    gemm_nt_wmma<<<g_dbc, blk, 0, stream>>>(hconv, D, W_dbc, D, nullptr, nullptr,
                                            dbcbuf, DBC, ML, DBC, D);
    // 6) delta_pre = dbc[:, :64] @ W_dt^T + b_dt   (softplus fused into scan)
    gemm_nt_wmma<<<g_nt, blk, 0, stream>>>(dbcbuf, DBC, W_dt, RANK, b_dt, nullptr,
                                           delta, D, ML, D, RANK);
    // 7) selective scan -> yssm (+ Dp * x term)
    ssm_scan_k<<<(Bn * D * NST) / 256, blk, 0, stream>>>(hconv, dbcbuf, delta,
                                                         A_log, Dp, yssm);
    // 8) ln2 = LayerNorm(yssm)
    layernorm_k<<<ML, blk, 0, stream>>>(yssm, ln_g, ln_b, ln2, D);
    // 9) hmm[b] = ln2[b] @ lskip[b]   (batched NN einsum 'bnd,bdk->bnk')
    gemm_nn_wmma<<<g_nn, blk, 0, stream>>>(ln2, D, 1, BAT, lskip, D, BAT, 0,
                                           nullptr, nullptr, 0, hmm, D, BAT, D, D, D);
    // 10) out = hmm @ Wp^T + bp + x
    gemm_nt_wmma<<<g_nt, blk, 0, stream>>>(hmm, D, Wp, D, bp, x, out, D, ML, D, D);
}